// PointPillarScatter_range_image_88545045774933
// MI455X (gfx1250) — compile-verified
//
#include <hip/hip_runtime.h>
#include <hip/hip_bf16.h>

// ---------------- problem constants (from the reference) ----------------
#define BATCH   2
#define P_PER   20000
#define NP_PER  60000
#define NXC     468
#define NYC     468
#define CFC     64
#define HRC     48
#define WRC     512
#define HWC     468                   // H == W == 468
#define PLANE   (468 * 468)           // 219024
#define OUTCH   128                   // 64 spatial + 64 bev
#define XOFF_C  248                   // -floor(Y0/RES)
#define YOFF_C  432                   // ceil(X1/RES)
#define XIMG_MAXC 495
#define YIMG_MAXC 431
#define PTS_PER_BLK 64                // points staged per block in bev kernel

// native clang vector (NOT HIP_vector_type) so nontemporal builtins accept it
typedef float v4f __attribute__((ext_vector_type(4)));

// the async-to-LDS builtin's payload type is a 16B int vector (per hipcc
// diagnostic: 'int __vector_size__(4*sizeof(int)) __device__ *')
typedef int v4i __attribute__((vector_size(16)));
typedef __attribute__((address_space(1))) v4i* gptr_b128;   // global (AS1)
typedef __attribute__((address_space(3))) v4i* lptr_b128;   // LDS    (AS3)

#if __has_builtin(__builtin_amdgcn_global_load_async_to_lds_b128)
#define HAVE_ASYNC_LDS 1
#else
#define HAVE_ASYNC_LDS 0
#endif

// 16B global -> LDS copy: async DMA path on gfx1250, sync fallback otherwise
__device__ __forceinline__ void lds_copy16(const void* g, void* s) {
#if HAVE_ASYNC_LDS
    __builtin_amdgcn_global_load_async_to_lds_b128(
        (gptr_b128)g, (lptr_b128)s, /*offset=*/0, /*cpol=*/0);
#else
    *(v4f*)s = *(const v4f*)g;
#endif
}

__device__ __forceinline__ void wait_async0() {
#if HAVE_ASYNC_LDS
#if __has_builtin(__builtin_amdgcn_s_wait_asynccnt)
    __builtin_amdgcn_s_wait_asynccnt(0);
#else
    asm volatile("s_wait_asynccnt 0x0" ::: "memory");
#endif
#endif
}

// ---------------- kernel: BEV gather+scale+scatter (channels 64..127) ---
// (defined FIRST so the disasm snippet shows the async-to-LDS staging)
// Block = 256 threads = (64 lanes = channel) x (4 point-groups), processing
// a 64-point chunk. Point records are staged into LDS with async b128 copies
// (ASYNCcnt) and then read back via same-address LDS broadcast, eliminating
// the 64x redundant global reads per point.
__global__ void __launch_bounds__(256)
bev_scatter(const float* __restrict__ feat_src, int transposed,
            const int*   __restrict__ lx,
            const int*   __restrict__ ly,
            const float* __restrict__ lp,
            float*       __restrict__ out) {
    __shared__ alignas(16) float s_lp[PTS_PER_BLK * 5];   // 1280 B
    __shared__ alignas(16) int   s_lx[PTS_PER_BLK * 2];   //  512 B
    __shared__ alignas(16) int   s_ly[PTS_PER_BLK * 2];   //  512 B

    const int  tid  = threadIdx.y * 64 + threadIdx.x;
    const long base = (long)blockIdx.x * PTS_PER_BLK;
    const long rem  = (long)BATCH * NP_PER - base;
    const int  npts = (rem >= PTS_PER_BLK) ? PTS_PER_BLK : (rem > 0 ? (int)rem : 0);
    if (npts <= 0) return;

    if (npts == PTS_PER_BLK) {
        // full chunk: 80 + 32 + 32 async b128 transfers (16B-aligned sources)
        if (tid < 80)
            lds_copy16((const char*)(lp + base * 5) + tid * 16,
                       (char*)s_lp + tid * 16);
        else if (tid < 112)
            lds_copy16((const char*)(lx + base * 2) + (tid - 80) * 16,
                       (char*)s_lx + (tid - 80) * 16);
        else if (tid < 144)
            lds_copy16((const char*)(ly + base * 2) + (tid - 112) * 16,
                       (char*)s_ly + (tid - 112) * 16);
        wait_async0();
    } else {
        for (int i = tid; i < npts * 5; i += 256) s_lp[i] = lp[base * 5 + i];
        for (int i = tid; i < npts * 2; i += 256) {
            s_lx[i] = lx[base * 2 + i];
            s_ly[i] = ly[base * 2 + i];
        }
    }
    __syncthreads();

    const int c = threadIdx.x;
    for (int i = threadIdx.y; i < npts; i += 4) {
        const long p = base + i;
        const int  b = (int)(p / NP_PER);

        float px = s_lp[i * 5 + 1];
        float py = s_lp[i * 5 + 2];
        float pz = s_lp[i * 5 + 3];

        // exact reference mask (incl. the y>=0 term that drops ~half the points)
        bool valid = (px > 0.0f)    && (px < 69.12f) &&
                     (py > -39.68f) && (py < 39.68f) &&
                     (px >= 0.0f)   && (px < 468.0f) &&
                     (py >= 0.0f)   && (py < 468.0f);
        if (!valid) continue;                        // reference: x_img=H -> dropped

        int x_img = (int)(-py / 0.16f) + XOFF_C;     // trunc-toward-zero == astype(int32)
        x_img = min(max(x_img, 0), XIMG_MAXC);
        int y_img = (int)(-px / 0.16f) + YOFF_C;
        y_img = min(max(y_img, 0), YIMG_MAXC);
        if (x_img >= HWC) continue;                  // mode='drop'

        float zc = fminf(fmaxf(pz, -2.0f), 4.0f);

        int xi = s_lx[i * 2 + 1];
        int yi = s_ly[i * 2 + 1];

        float f;
        if (transposed)
            f = feat_src[(((long)b * HRC + yi) * WRC + xi) * CFC + c]; // 256B/point, coalesced
        else
            f = feat_src[(((long)b * CFC + c) * HRC + yi) * WRC + xi]; // fallback (L2 hits)

        // bev[c, y_img, x_img] = im[x_img, y_img, c] -> out[b, 64+c, y_img, x_img]
        long off = (((long)b * OUTCH + 64 + c) * HWC + y_img) * HWC + x_img;
        __builtin_nontemporal_store(zc * f, out + off);
    }
}

// ---------------- kernel: streaming zero-fill of d_out ------------------
// 224 MB of pure streaming stores -> b128 non-temporal so we don't flush
// the 192 MB L2 (range_out / laser tables / BEV RMW region stay resident).
__global__ void zero_f4_nt(v4f* __restrict__ p, long n4) {
    long i = (long)blockIdx.x * blockDim.x + threadIdx.x;
    if (i >= n4) return;
    v4f z = (v4f)(0.0f);
    __builtin_nontemporal_store(z, p + i);
}

__global__ void zero_tail(float* __restrict__ p, int n) {
    int i = threadIdx.x;
    if (i < n) __builtin_nontemporal_store(0.0f, p + i);
}

// ---------------- kernel: transpose range_out (B,CF,HR,WR)->(B,HR,WR,CF)
// c is fastest in thread order -> coalesced b32 writes; source is 12.6 MB,
// read once (L2-resident). After this, each point's 64-channel gather is a
// single contiguous 256B read, perfectly coalesced across the 64 lanes.
__global__ void transpose_rc(const float* __restrict__ ro, float* __restrict__ rt) {
    long t = (long)blockIdx.x * blockDim.x + threadIdx.x;
    const long total = (long)BATCH * CFC * HRC * WRC;
    if (t >= total) return;
    int  c    = (int)(t & (CFC - 1));
    long rest = t >> 6;                 // (b*HR + y)*WR + x
    int  x = (int)(rest % WRC); rest /= WRC;
    int  y = (int)(rest % HRC);
    int  b = (int)(rest / HRC);
    rt[t] = ro[(((long)b * CFC + c) * HRC + y) * WRC + x];
}

// ---------------- kernel: pillar scatter (channels 0..63) ---------------
__global__ void pillar_scatter(const float* __restrict__ pf,
                               const int*   __restrict__ vc,
                               float*       __restrict__ out) {
    int t = blockIdx.x * blockDim.x + threadIdx.x;
    int p = t >> 6;
    int c = t & 63;
    if (p >= BATCH * P_PER) return;
    __builtin_prefetch(vc + (p + 256) * 4, 0, 1);   // global_prefetch_b8
    int b  = vc[p * 4 + 0];
    int zz = vc[p * 4 + 1];
    int cy = vc[p * 4 + 2];
    int cx = vc[p * 4 + 3];
    int lin = zz + cy * NXC + cx;                   // reference: c1 + c2*NX + c3
    float v = pf[p * CFC + c];
    long off = ((long)b * OUTCH + c) * PLANE + lin;
    __builtin_nontemporal_store(v, out + off);
}

// ---------------- launch ------------------------------------------------
extern "C" void kernel_launch(void* const* d_in, const int* in_sizes, int n_in,
                              void* d_out, int out_size, void* d_ws, size_t ws_size,
                              hipStream_t stream) {
    const float* pf = (const float*)d_in[0];   // pillar_features (B*P_PER, 64)
    const int*   vc = (const int*)  d_in[1];   // voxel_coords    (B*P_PER, 4)
    const float* ro = (const float*)d_in[2];   // range_out       (B, 64, 48, 512)
    const int*   lx = (const int*)  d_in[3];   // laser_x         (B*NP_PER, 2)
    const int*   ly = (const int*)  d_in[4];   // laser_y         (B*NP_PER, 2)
    const float* lp = (const float*)d_in[5];   // laser_points    (B*NP_PER, 5)
    float* out = (float*)d_out;

    // 1) zero-fill output (dominant cost, ~224 MB streaming NT b128)
    long n  = (long)out_size;
    long n4 = n >> 2;
    int  zblocks = (int)((n4 + 255) / 256);
    zero_f4_nt<<<zblocks, 256, 0, stream>>>((v4f*)out, n4);
    if (n & 3)
        zero_tail<<<1, 64, 0, stream>>>(out + (n4 << 2), (int)(n & 3));

    // 2) optional channel-major transpose of range_out into workspace
    const size_t rt_bytes = (size_t)BATCH * HRC * WRC * CFC * sizeof(float);
    bool   use_t = (ws_size >= rt_bytes) && (d_ws != nullptr);
    float* rt    = (float*)d_ws;
    if (use_t) {
        long tt = (long)BATCH * CFC * HRC * WRC;
        transpose_rc<<<(int)((tt + 255) / 256), 256, 0, stream>>>(ro, rt);
    }

    // 3) pillar scatter into channels 0..63
    int pt = BATCH * P_PER * CFC;
    pillar_scatter<<<(pt + 255) / 256, 256, 0, stream>>>(pf, vc, out);

    // 4) BEV gather+scatter into channels 64..127 (async-LDS staged chunks)
    dim3 bt(64, 4);
    int  nb = (int)(((long)BATCH * NP_PER + PTS_PER_BLK - 1) / PTS_PER_BLK);
    bev_scatter<<<nb, bt, 0, stream>>>(use_t ? rt : ro, use_t ? 1 : 0,
                                       lx, ly, lp, out);
}